// get_histogram_10995116278399
// MI455X (gfx1250) — compile-verified
//
#include <hip/hip_runtime.h>
#include <stdint.h>

// ---------------------------------------------------------------------------
// Histogram of 64x3x512x512 fp32 image batch.
// out = [hist_r(256), bins(257), count_r(256), count_g(256), count_b(256)] = 1281 f32
// ws (u32): [0..767] per-channel counts (c*256+bin), [768..1023] hist_r counts
// Memory-bound: 201 MB @ 23.3 TB/s ~= 8.6us floor. Strategy: b128 loads,
// wave32-private LDS histograms (ds_add_u32), u32 global atomics, f32 finalize.
// k_hist_r additionally demonstrates the CDNA5 async global->LDS path.
// ---------------------------------------------------------------------------

#define NBINS 256
#define HW    (512 * 512)
#define HW4   (HW / 4)

#if defined(__has_builtin)
#if __has_builtin(__builtin_amdgcn_global_load_async_to_lds_b128) && \
    __has_builtin(__builtin_amdgcn_s_wait_asynccnt)
#define HAVE_ASYNC_LDS 1
#endif
#endif

#if defined(HAVE_ASYNC_LDS)
typedef int v4i __attribute__((vector_size(16)));
typedef __attribute__((address_space(1))) v4i* g4p_t;
typedef __attribute__((address_space(3))) v4i* l4p_t;
#endif

__device__ __forceinline__ int bin_of(float x) {
    // reference: idx = clip(floor((x*255) * (256/255)), 0, 255)
    float v = (x * 255.0f) * (256.0f / 255.0f);
    int b = (int)floorf(v);
    b = b < 0 ? 0 : b;
    b = b > 255 ? 255 : b;
    return b;
}

__global__ void k_init(unsigned* __restrict__ ws) {
    int i = blockIdx.x * blockDim.x + threadIdx.x;
    if (i < 1024) ws[i] = 0u;
}

// Main 3-channel histogram over the whole batch (bandwidth-critical pass:
// direct b128 loads, no LDS staging round-trip).
__global__ void __launch_bounds__(256) k_hist(const float* __restrict__ in,
                                              const int* __restrict__ pbatch,
                                              unsigned* __restrict__ counts) {
    constexpr int WAVES = 8;
    constexpr int RST   = 3 * NBINS + 16;   // pad to stagger LDS banks per wave
    __shared__ unsigned lh[WAVES * RST];
    for (int i = threadIdx.x; i < WAVES * RST; i += 256) lh[i] = 0u;
    __syncthreads();

    const int  b   = *pbatch;
    const long n4  = (long)b * 3 * HW4;     // float4 element count
    unsigned*  mh  = &lh[(threadIdx.x >> 5) * RST];
    const float4* in4 = (const float4*)in;
    const long stride = (long)gridDim.x * 256;

    for (long i = (long)blockIdx.x * 256 + threadIdx.x; i < n4; i += stride) {
        __builtin_prefetch(&in4[i + stride], 0, 0);   // global_prefetch_b8
        float4 v = in4[i];
        int c = (int)((i >> 16) % 3);                 // HW4 == 65536
        unsigned* ch = mh + (c << 8);
        atomicAdd(&ch[bin_of(v.x)], 1u);
        atomicAdd(&ch[bin_of(v.y)], 1u);
        atomicAdd(&ch[bin_of(v.z)], 1u);
        atomicAdd(&ch[bin_of(v.w)], 1u);
    }
    __syncthreads();

    for (int i = threadIdx.x; i < 3 * NBINS; i += 256) {
        unsigned s = 0;
#pragma unroll
        for (int w = 0; w < WAVES; ++w) s += lh[w * RST + i];
        if (s) atomicAdd(&counts[i], s);
    }
}

// hist_r: channel 0 of the LAST batch item (loop-leak semantics). 1 MB total.
// 256 blocks x 256 threads == HW4 threads: one float4 per thread, staged into
// LDS via CDNA5 async global->LDS DMA when available.
__global__ void __launch_bounds__(256) k_hist_r(const float* __restrict__ in,
                                                const int* __restrict__ pbatch,
                                                unsigned* __restrict__ histr) {
    constexpr int WAVES = 8;
    constexpr int RST   = NBINS + 16;
    __shared__ unsigned lh[WAVES * RST];
    for (int i = threadIdx.x; i < WAVES * RST; i += 256) lh[i] = 0u;
    __syncthreads();

    const int b = *pbatch;
    const float4* in4 = (const float4*)in + (long)(b - 1) * 3 * HW4;
    unsigned* mh = &lh[(threadIdx.x >> 5) * RST];
    const int i = blockIdx.x * 256 + threadIdx.x;

    if (i < HW4) {
#if defined(HAVE_ASYNC_LDS)
        __shared__ float4 stage[256];
        // Per-lane async DMA: global b128 -> this wave's LDS slots.
        g4p_t gp = (g4p_t)(uintptr_t)(&in4[i]);
        l4p_t lp = (l4p_t)(uintptr_t)(&stage[threadIdx.x]);
        __builtin_amdgcn_global_load_async_to_lds_b128(gp, lp, 0, 0);
        __builtin_amdgcn_s_wait_asynccnt(0);   // wave reads only its own slots
        __syncthreads();
        float4 v = stage[threadIdx.x];
#else
        float4 v = in4[i];
#endif
        atomicAdd(&mh[bin_of(v.x)], 1u);
        atomicAdd(&mh[bin_of(v.y)], 1u);
        atomicAdd(&mh[bin_of(v.z)], 1u);
        atomicAdd(&mh[bin_of(v.w)], 1u);
    }
    __syncthreads();

    for (int i2 = threadIdx.x; i2 < NBINS; i2 += 256) {
        unsigned s = 0;
#pragma unroll
        for (int w = 0; w < WAVES; ++w) s += lh[w * RST + i2];
        if (s) atomicAdd(&histr[i2], s);
    }
}

__global__ void k_final(const unsigned* __restrict__ ws, float* __restrict__ out) {
    int i = blockIdx.x * blockDim.x + threadIdx.x;
    if (i >= 1281) return;
    if (i < 256) {
        out[i] = (float)ws[768 + i];                       // hist_r
    } else if (i < 513) {
        out[i] = (float)(i - 256) * (255.0f / 256.0f);     // bins (exact dyadic step)
    } else {
        out[i] = (float)ws[i - 513];                       // count_r/g/b
    }
}

// ---------------------------------------------------------------------------

extern "C" void kernel_launch(void* const* d_in, const int* in_sizes, int n_in,
                              void* d_out, int out_size, void* d_ws, size_t ws_size,
                              hipStream_t stream) {
    const int*   pbatch = (const int*)d_in[0];     // device scalar batchsize
    const float* img    = (const float*)d_in[1];   // [B,3,512,512] fp32
    unsigned*    ws     = (unsigned*)d_ws;
    float*       out    = (float*)d_out;

    k_init  <<<4,    256, 0, stream>>>(ws);
    k_hist  <<<4096, 256, 0, stream>>>(img, pbatch, ws);
    k_hist_r<<<256,  256, 0, stream>>>(img, pbatch, ws + 768);
    k_final <<<6,    256, 0, stream>>>(ws, out);
}